// PeLayer_11458972745980
// MI455X (gfx1250) — compile-verified
//
#include <hip/hip_runtime.h>

// ---------------------------------------------------------------------------
// PE-bias add:  out[b,h,i,j] = scores[b,h,i,j] + W1[rp(i,j), bmat(b,i,j), h]
//
// Pure bandwidth problem: 1 GiB of HBM traffic -> ~46us floor @ 23.3 TB/s.
// No matmul => no WMMA. CDNA5 features used: async global->LDS staging of
// the 130KB W1 gather table (ASYNCcnt + s_wait_asynccnt), DS-B128 bias
// gathers, and B128 non-temporal streaming of scores/out.
// ---------------------------------------------------------------------------

#define SEQ_L    2048
#define N_HEAD   16
#define N_MB     8
#define NUM_BK   32
#define W1_ROWS  (N_MB * N_MB + 1)            // 65
#define W1_FLOATS (NUM_BK * W1_ROWS * N_HEAD) // 33280 floats = 130 KiB
#define BATCH    2
#define ROWS_PER_BLOCK 4                      // 1024 blocks: fills a big chip
#define THREADS  256

typedef float v4f __attribute__((ext_vector_type(4)));
typedef int   v4i __attribute__((ext_vector_type(4)));

// ROCm signature: (int4 addrspace(1)*, int4 addrspace(3)*, imm offset, imm cpol)
typedef __attribute__((address_space(1))) v4i* gv4i_ptr;
typedef __attribute__((address_space(3))) v4i* lv4i_ptr;

#if defined(__has_builtin)
#  if __has_builtin(__builtin_amdgcn_global_load_async_to_lds_b128)
#    define USE_ASYNC_LDS 1
#  endif
#endif

__global__ __launch_bounds__(THREADS) void pe_bias_add_kernel(
    const float* __restrict__ scores,   // [B, H, L, L]
    const int*   __restrict__ bseq,     // [B, L]
    const float* __restrict__ w1,       // [32, 65, 16]
    float*       __restrict__ out)      // [B, H, L, L]
{
    extern __shared__ char smem_raw[];
    float* w1s  = (float*)smem_raw;                                  // 33280 f32
    int*   seqs = (int*)(smem_raw + W1_FLOATS * sizeof(float));      // 2048 i32

    const int tid  = threadIdx.x;
    const int row0 = blockIdx.x * ROWS_PER_BLOCK;     // flat (b,i) row group
    const int b    = row0 / SEQ_L;                    // rows in block share b
    const int i0   = row0 % SEQ_L;

    // ---- Stage W1 into LDS: CDNA5 async global->LDS (ASYNCcnt) ----------
    const int NV4 = W1_FLOATS / 4;                    // 8320 float4s
    for (int idx = tid; idx < NV4; idx += THREADS) {
#ifdef USE_ASYNC_LDS
        __builtin_amdgcn_global_load_async_to_lds_b128(
            (gv4i_ptr)(w1  + 4 * idx),
            (lv4i_ptr)(w1s + 4 * idx),
            /*offset=*/0, /*cpol=*/0);
#else
        ((v4f*)w1s)[idx] = ((const v4f*)w1)[idx];
#endif
    }
    // Stage this batch's b_seq row (8 KiB)
    for (int idx = tid; idx < SEQ_L; idx += THREADS)
        seqs[idx] = bseq[b * SEQ_L + idx];

#ifdef USE_ASYNC_LDS
#  if defined(__has_builtin) && __has_builtin(__builtin_amdgcn_s_wait_asynccnt)
    __builtin_amdgcn_s_wait_asynccnt(0);
#  else
    asm volatile("s_wait_asynccnt 0" ::: "memory");
#  endif
#endif
    __syncthreads();

    const size_t LL  = (size_t)SEQ_L * SEQ_L;   // 4M elements per head plane
    const size_t LL4 = LL / 4;                  // in float4 units
    const size_t bhBase = (size_t)b * N_HEAD * LL;

    for (int r = 0; r < ROWS_PER_BLOCK; ++r) {
        const int i  = i0 + r;
        const int bi = seqs[i];

        // 2048 j's / (256 threads * 4-wide) = 2 chunks per row
        for (int c = 0; c < SEQ_L / (THREADS * 4); ++c) {
            const int j0 = c * (THREADS * 4) + tid * 4;

            // per-j bias base index into W1 (16 head values contiguous)
            int base[4];
#pragma unroll
            for (int k = 0; k < 4; ++k) {
                const int j  = j0 + k;
                const int bj = seqs[j];
                const int bm = (bi != 0 && bj != 0) ? ((bi - 1) * N_MB + bj) : 0;

                // relative_position_bucket(j - i):  n = i - j
                int n = i - j;
                int ret = 0;
                if (n < 0) { ret = 16; n = -n; }
                int rp;
                if (n < 8) {
                    rp = ret + n;
                } else {
                    // 8 + trunc( log(n/8) / log(16) * 8 ), clamped to 15
                    int v = 8 + (int)(logf((float)n / 8.0f)
                                      / 2.772588722239781f * 8.0f);
                    if (v > 15) v = 15;
                    rp = ret + v;
                }
                base[k] = (rp * W1_ROWS + bm) * N_HEAD;   // *4B = 64B aligned
            }

            const size_t e0 = bhBase + (size_t)i * SEQ_L + j0;  // h = 0 offset
            const v4f* in4  = (const v4f*)(scores + e0);
            v4f*       out4 = (v4f*)(out + e0);

            // 16 heads in two half-groups of 8: batch 8 NT B128 loads in
            // flight before use, then bias-add from LDS, then 8 NT stores.
#pragma unroll
            for (int half = 0; half < 2; ++half) {
                const int h0 = half * 8;

                v4f s[8];
#pragma unroll
                for (int hh = 0; hh < 8; ++hh)
                    s[hh] = __builtin_nontemporal_load(
                        in4 + (size_t)(h0 + hh) * LL4);

                v4f bb[4][2];               // bias[h0..h0+7] per j-lane k
#pragma unroll
                for (int k = 0; k < 4; ++k) {
                    bb[k][0] = *(const v4f*)(w1s + base[k] + h0);
                    bb[k][1] = *(const v4f*)(w1s + base[k] + h0 + 4);
                }

#pragma unroll
                for (int hh = 0; hh < 8; ++hh) {
                    v4f t = s[hh];
                    t.x += bb[0][hh >> 2][hh & 3];
                    t.y += bb[1][hh >> 2][hh & 3];
                    t.z += bb[2][hh >> 2][hh & 3];
                    t.w += bb[3][hh >> 2][hh & 3];
                    __builtin_nontemporal_store(t, out4 + (size_t)(h0 + hh) * LL4);
                }
            }
        }
    }
}

extern "C" void kernel_launch(void* const* d_in, const int* in_sizes, int n_in,
                              void* d_out, int out_size, void* d_ws, size_t ws_size,
                              hipStream_t stream) {
    const float* scores = (const float*)d_in[0];   // (2,16,2048,2048) f32
    const int*   bseq   = (const int*)d_in[1];     // (2,2048) i32
    const float* w1     = (const float*)d_in[2];   // (32,65,16) f32
    float*       out    = (float*)d_out;

    const int    blocks = (BATCH * SEQ_L) / ROWS_PER_BLOCK;              // 1024
    const size_t shmem  = W1_FLOATS * sizeof(float) + SEQ_L * sizeof(int); // 141312 B

    pe_bias_add_kernel<<<blocks, THREADS, shmem, stream>>>(scores, bseq, w1, out);
}